// ConvectionDilationPdeM2_18880676233735
// MI455X (gfx1250) — compile-verified
//
#include <hip/hip_runtime.h>
#include <math.h>
#include <stdint.h>

// Problem geometry (fixed by the reference).
#define C_   16
#define OR_  8
#define H_   192
#define W_   192
#define VOL  (C_*OR_*H_*W_)                 // 4,718,592 f32 = 18.9 MB
#define TWO_PI_F 6.283185307179586f
#define DTH  (TWO_PI_F / (float)OR_)

// Dilation tile geometry: 32x8 outputs per block, halo radius 2
// (|rotated offset| <= sqrt(2), bilinear adds +1).
#define TX 32
#define TY 8
#define LW (TX + 4)       // 36 valid columns
#define LH (TY + 4)       // 12 rows
#define LSTRIDE 37        // padded stride (bank-friendly)

// ---------------------------------------------------------------------------
// CDNA5 async global->LDS copy (ASYNCcnt path). Falls back to a plain
// load+ds_store when not compiling for gfx1250.
// ---------------------------------------------------------------------------
typedef __attribute__((address_space(1))) int GInt;   // global
typedef __attribute__((address_space(3))) int LInt;   // LDS

__device__ __forceinline__ void async_tile_b32(const float* src, float* dst_lds) {
#if defined(__gfx1250__)
#if __has_builtin(__builtin_amdgcn_global_load_async_to_lds_b32)
    __builtin_amdgcn_global_load_async_to_lds_b32(
        (GInt*)src, (LInt*)dst_lds, 0, 0);
#else
    unsigned lds_off = (unsigned)(uintptr_t)(LInt*)dst_lds;
    asm volatile("global_load_async_to_lds_b32 %0, %1, off"
                 :: "v"(lds_off), "v"(src) : "memory");
#endif
#else
    *dst_lds = *src;
#endif
}

__device__ __forceinline__ void wait_async_lds() {
#if defined(__gfx1250__)
#if __has_builtin(__builtin_amdgcn_s_wait_asynccnt)
    __builtin_amdgcn_s_wait_asynccnt(0);
#else
    asm volatile("s_wait_asynccnt 0" ::: "memory");
#endif
#endif
}

__device__ __forceinline__ int iclamp(int v, int lo, int hi) {
    return v < lo ? lo : (v > hi ? hi : v);
}

// ---------------------------------------------------------------------------
// Setup: kcost[c][27] table and per-(c,t) convection parameters.
// cp[(c*8+t)*8] = { floor(sx), floor(sy), wx, wy, wt, t0i, 0, 0 }
// ---------------------------------------------------------------------------
__global__ __launch_bounds__(512) void setup_kernel(const float* __restrict__ g0,
                                                    const float* __restrict__ metric,
                                                    float* __restrict__ kc,
                                                    float* __restrict__ cp) {
    const int tid = threadIdx.x;
    if (tid < C_ * 27) {
        const int c = tid / 27, j = tid % 27;
        const int ht = j / 9 - 1;
        const int hy = (j / 3) % 3 - 1;
        const int hx = j % 3 - 1;
        const float hth  = (float)ht * DTH;
        const float half = 0.5f * hth;
        const float q = (fabsf(half) < 1e-4f) ? (1.0f - half * half * (1.0f / 3.0f))
                                              : (half / tanf(half));
        const float c1 =  q    * (float)hx + half * (float)hy;
        const float c2 = -half * (float)hx + q    * (float)hy;
        const float c3 = hth;
        const float w0 = metric[c * 3 + 0];
        const float w1 = metric[c * 3 + 1];
        const float w2 = metric[c * 3 + 2];
        const float a0 = w0 * c1, a1 = w1 * c2, a2 = w2 * c3;
        const float d2 = a0 * a0 + a1 * a1 + a2 * a2;
        const float ee = 2.0f * 0.65f / (2.0f * 0.65f - 1.0f);   // 4.3333335
        const float nu = (2.0f * 0.65f - 1.0f) * powf(2.0f * 0.65f, -ee);
        kc[tid] = nu * powf(d2, 0.5f * ee);
    }
    if (tid < C_ * OR_) {
        const int c = tid >> 3, t = tid & 7;
        const float x0  = g0[c * 3 + 0];
        const float y0  = g0[c * 3 + 1];
        const float th0 = g0[c * 3 + 2];
        const float a  = (float)t * DTH - th0;
        const float ca = cosf(a), sa = sinf(a);
        const float sx = -(ca * x0 - sa * y0);
        const float sy = -(sa * x0 + ca * y0);
        const float fx0 = floorf(sx), fy0 = floorf(sy);
        float tc = fmodf((float)t - th0 * ((float)OR_ / TWO_PI_F), (float)OR_);
        if (tc < 0.0f) tc += (float)OR_;
        const float t0f = floorf(tc);
        const int   t0i = ((int)t0f) & 7;
        float* e = cp + (size_t)tid * 8;
        e[0] = fx0; e[1] = fy0;
        e[2] = sx - fx0; e[3] = sy - fy0;
        e[4] = tc - t0f; e[5] = (float)t0i;
        e[6] = 0.0f; e[7] = 0.0f;
    }
}

// ---------------------------------------------------------------------------
// Convection: per (c,t)-plane spatially-uniform trilinear resample.
// 8 coalesced taps per output; weights / integer shifts are lane-uniform.
// ---------------------------------------------------------------------------
__global__ __launch_bounds__(W_) void conv_kernel(const float* __restrict__ u,
                                                  const float* __restrict__ cp,
                                                  float* __restrict__ out) {
    const int x  = threadIdx.x;       // 0..191
    const int y  = blockIdx.x;        // 0..191
    const int ct = blockIdx.y;        // 0..127 (c*8+t)
    const int c  = ct >> 3;

    const float* e = cp + (size_t)ct * 8;
    const int   ix  = x + (int)e[0];
    const int   iy  = y + (int)e[1];
    const float wx  = e[2], wy = e[3], wt = e[4];
    const int   t0i = (int)e[5];
    const int   t1i = (t0i + 1) & 7;

    const float mx0 = (ix     >= 0 && ix     < W_) ? 1.0f : 0.0f;
    const float mx1 = (ix + 1 >= 0 && ix + 1 < W_) ? 1.0f : 0.0f;
    const float my0 = (iy     >= 0 && iy     < H_) ? 1.0f : 0.0f;
    const float my1 = (iy + 1 >= 0 && iy + 1 < H_) ? 1.0f : 0.0f;
    const int ix0 = iclamp(ix,     0, W_ - 1);
    const int ix1 = iclamp(ix + 1, 0, W_ - 1);
    const int iy0 = iclamp(iy,     0, H_ - 1);
    const int iy1 = iclamp(iy + 1, 0, H_ - 1);

    const float* b0 = u + ((size_t)(c * OR_ + t0i)) * (H_ * W_);
    const float* b1 = u + ((size_t)(c * OR_ + t1i)) * (H_ * W_);

    const float w00 = (1.0f - wy) * (1.0f - wx);
    const float w01 = (1.0f - wy) * wx;
    const float w10 = wy * (1.0f - wx);
    const float w11 = wy * wx;

    const float s0 = w00 * (b0[iy0 * W_ + ix0] * (my0 * mx0))
                   + w01 * (b0[iy0 * W_ + ix1] * (my0 * mx1))
                   + w10 * (b0[iy1 * W_ + ix0] * (my1 * mx0))
                   + w11 * (b0[iy1 * W_ + ix1] * (my1 * mx1));
    const float s1 = w00 * (b1[iy0 * W_ + ix0] * (my0 * mx0))
                   + w01 * (b1[iy0 * W_ + ix1] * (my0 * mx1))
                   + w10 * (b1[iy1 * W_ + ix0] * (my1 * mx0))
                   + w11 * (b1[iy1 * W_ + ix1] * (my1 * mx1));

    out[((size_t)ct * H_ + y) * W_ + x] = (1.0f - wt) * s0 + wt * s1;
}

// ---------------------------------------------------------------------------
// Fractional dilation: max over 27 offsets of bilinear(u) - kcost.
// 3 orientation planes of a halo tile staged in LDS via async loads (ASYNCcnt);
// zero-filled halo == reference's zero padding. All bilinear weights and
// integer tap offsets are lane-uniform, and the per-block scalar setup
// (cos/sin, 9 weight tuples, kcost prefetch) overlaps the async drain.
// ---------------------------------------------------------------------------
__global__ __launch_bounds__(256) void dil_kernel(const float* __restrict__ u,
                                                  const float* __restrict__ kc,
                                                  float* __restrict__ out) {
    __shared__ float tile[3 * LH * LSTRIDE];

    const int tid   = threadIdx.x;
    const int lx    = tid & 31;
    const int ly    = tid >> 5;
    const int tileX = blockIdx.x * TX;
    const int tileY = blockIdx.y * TY;
    const int ctid  = blockIdx.z;           // c*8+t
    const int c     = ctid >> 3;
    const int t     = ctid & 7;
    const size_t chanBase = (size_t)c * OR_ * H_ * W_;

    // Stage planes (t-1, t, t+1) mod 8, halo radius 2, zeros outside domain.
    for (int i = tid; i < 3 * LH * LW; i += 256) {
        const int p   = i / (LH * LW);
        const int rem = i - p * (LH * LW);
        const int r   = rem / LW;
        const int col = rem - r * LW;
        const int gy  = tileY - 2 + r;
        const int gx  = tileX - 2 + col;
        const int tp  = (t + p + 7) & 7;
        float* dst = &tile[(p * LH + r) * LSTRIDE + col];
        if ((unsigned)gy < (unsigned)H_ && (unsigned)gx < (unsigned)W_) {
            async_tile_b32(u + chanBase + ((size_t)(tp * H_ + gy)) * W_ + gx, dst);
        } else {
            *dst = 0.0f;
        }
    }

    // --- Scalar setup overlapped with the async copies -------------------
    const float theta = (float)t * DTH;
    const float ctc = cosf(theta), stc = sinf(theta);
    const float* kcc = kc + c * 27;

    int   ixo[9], iyo[9];
    float wxa[9], wya[9];
    float kcr[27];
#pragma unroll
    for (int yy = 0; yy < 3; ++yy) {
#pragma unroll
        for (int xx = 0; xx < 3; ++xx) {
            const int s = yy * 3 + xx;
            const float hx = (float)(xx - 1);
            const float hy = (float)(yy - 1);
            const float fx = ctc * hx - stc * hy;     // lane-uniform
            const float fy = stc * hx + ctc * hy;
            const float bx = floorf(fx), by = floorf(fy);
            wxa[s] = fx - bx;
            wya[s] = fy - by;
            ixo[s] = (int)bx;                         // in [-2,1]
            iyo[s] = (int)by;
        }
    }
#pragma unroll
    for (int j = 0; j < 27; ++j) kcr[j] = kcc[j];

    wait_async_lds();
    __syncthreads();

    float best = -INFINITY;
#pragma unroll
    for (int p = 0; p < 3; ++p) {
        const float* tp_ = tile + p * LH * LSTRIDE;
#pragma unroll
        for (int s = 0; s < 9; ++s) {
            const float wx = wxa[s], wy = wya[s];
            const int ix = lx + 2 + ixo[s];           // in [0,34]
            const int iy = ly + 2 + iyo[s];           // in [0,10]
            const float* q = tp_ + iy * LSTRIDE + ix;
            const float v00 = q[0];
            const float v01 = q[1];
            const float v10 = q[LSTRIDE];
            const float v11 = q[LSTRIDE + 1];
            const float v = (1.0f - wy) * (1.0f - wx) * v00
                          + (1.0f - wy) * wx          * v01
                          + wy          * (1.0f - wx) * v10
                          + wy          * wx          * v11;
            best = fmaxf(best, v - kcr[p * 9 + s]);
        }
    }

    out[((size_t)ctid * H_ + (tileY + ly)) * W_ + (tileX + lx)] = best;
}

// ---------------------------------------------------------------------------
// Launch: setup, then 2x (convection -> dilation). d_out itself is used as
// the second ping-pong field, so only one scratch field is required.
// ---------------------------------------------------------------------------
extern "C" void kernel_launch(void* const* d_in, const int* in_sizes, int n_in,
                              void* d_out, int out_size, void* d_ws, size_t ws_size,
                              hipStream_t stream) {
    const float* u_in   = (const float*)d_in[0];   // [1,16,8,192,192]
    const float* g0     = (const float*)d_in[1];   // [16,3]
    const float* metric = (const float*)d_in[2];   // [16,3]
    float* out = (float*)d_out;

    char* ws = (char*)d_ws;
    float* kc   = (float*)ws;                 // 432 floats
    float* cp   = kc + 512;                   // 1024 floats
    float* bufA = (float*)(ws + 8192);        // VOL floats (only scratch field)

    setup_kernel<<<dim3(1), dim3(512), 0, stream>>>(g0, metric, kc, cp);

    const dim3 convGrid(H_, C_ * OR_);
    const dim3 convBlk(W_);
    const dim3 dilGrid(W_ / TX, H_ / TY, C_ * OR_);
    const dim3 dilBlk(256);

    // Iteration 1
    conv_kernel<<<convGrid, convBlk, 0, stream>>>(u_in, cp, bufA);
    dil_kernel <<<dilGrid,  dilBlk,  0, stream>>>(bufA, kc, out);
    // Iteration 2
    conv_kernel<<<convGrid, convBlk, 0, stream>>>(out, cp, bufA);
    dil_kernel <<<dilGrid,  dilBlk,  0, stream>>>(bufA, kc, out);
}